// Loss_Longtail_46505905881691
// MI455X (gfx1250) — compile-verified
//
#include <hip/hip_runtime.h>
#include <hip/hip_bf16.h>

typedef __attribute__((ext_vector_type(2))) float v2f;
typedef __attribute__((ext_vector_type(8))) float v8f;

// One block (256 threads = 8 wave32) per batch row.
// Phase 1: lanes 0..L-1 gather tail scores into LDS.
// Phase 2: whole block streams the 32000-float row with b128 loads, per-lane
//          exp-accumulate (TRANS pipe), prefetch ahead.
// Phase 3: 8-wave partials folded to 32 lanes, then reduced with
//          v_wmma_f32_16x16x4_f32 (A = partials, B = ones).
// Phase 4: thread 0 runs the 50-step Plackett-Luce reverse scan.
__global__ __launch_bounds__(256)
void Loss_Longtail_46505905881691_kernel(const float* __restrict__ output,
                                         const int*   __restrict__ target,
                                         const int*   __restrict__ longtail,
                                         const float* __restrict__ loss_weight,
                                         float*       __restrict__ out,
                                         int V, int L) {
    const int b   = blockIdx.x;
    const int tid = threadIdx.x;

    __shared__ float sh_s[64];
    __shared__ float sh_es[64];
    __shared__ int   sh_m[64];
    __shared__ float red[256];
    __shared__ float sh_total;

    const float* row = output + (size_t)b * (size_t)V;

    // ---- Phase 1: parallel gather of ragged tail scores ----
    if (tid < L) {
        int idx = longtail[b * L + tid];
        float s = 0.0f, es = 0.0f;
        int m = (idx > 0) ? 1 : 0;
        if (m) {
            int ci = idx < V ? idx : (V - 1);
            s  = row[ci];
            es = __expf(s);
        }
        sh_s[tid]  = s;
        sh_es[tid] = es;
        sh_m[tid]  = m;
    }

    // ---- Phase 2: streaming row reduction sum(exp(x)) ----
    const int V4 = V >> 2;                       // V multiple of 4 (32000)
    const float4* row4 = (const float4*)row;
    float acc = 0.0f;
    for (int j = tid; j < V4; j += 256) {
        // keep the HBM stream deep: prefetch ~2K elements (512 vec4) ahead
        if (j + 512 < V4)
            __builtin_prefetch((const void*)(row4 + j + 512), 0, 1);
        float4 v = row4[j];
        acc += __expf(v.x) + __expf(v.y) + __expf(v.z) + __expf(v.w);
    }
    red[tid] = acc;
    __syncthreads();

    // ---- Phase 3: block reduction; final 32->1 via WMMA ----
    if (tid < 32) {
        float p = 0.0f;
#pragma unroll
        for (int i = 0; i < 8; ++i) p += red[tid + 32 * i];

        // A-matrix (16x4 f32, 2 VGPRs): VGPR0 lanes0-15 -> A[m,0]=p[m],
        // lanes16-31 -> A[m,2]=p[m+16]; VGPR1 = 0 -> A[m,1]=A[m,3]=0.
        // B = all ones (layout independent). D[m,n] = p[m] + p[m+16].
        v2f a;  a[0] = p;    a[1] = 0.0f;
        v2f bb; bb[0] = 1.0f; bb[1] = 1.0f;
        v8f c = {};
        v8f d = __builtin_amdgcn_wmma_f32_16x16x4_f32(
            /*neg_a=*/false, a, /*neg_b=*/false, bb,
            /*c_mod=*/(short)0, c, /*reuse_a=*/false, /*reuse_b=*/false);

        // Per lane: sum of its 8 D rows. lane<16 holds sum over M=0..7,
        // lane>=16 holds sum over M=8..15 (any column N).
        float t = d[0] + d[1] + d[2] + d[3] + d[4] + d[5] + d[6] + d[7];
        float total = __shfl(t, 0, 32) + __shfl(t, 16, 32);
        if (tid == 0) sh_total = total;
    }
    __syncthreads();

    // ---- Phase 4: serial tail scan (cheap: <= 50 steps) ----
    if (tid == 0) {
        float sum_exp = sh_total;
        int   tgt = target[b];
        float cw  = loss_weight[tgt];
        float ts  = row[tgt];
        float log_pl = ts - __logf(sum_exp);

        float tail_mass = 0.0f;
        for (int l = 0; l < L; ++l) tail_mass += sh_es[l];
        float other = sum_exp - __expf(ts) - tail_mass;

        float running = other;
        float tail = 0.0f;
        for (int l = L - 1; l >= 0; --l) {
            running += sh_es[l];
            if (sh_m[l]) tail += sh_s[l] - __logf(running);
        }
        out[b] = -(log_pl + tail) + cw;
    }
}

extern "C" void kernel_launch(void* const* d_in, const int* in_sizes, int n_in,
                              void* d_out, int out_size, void* d_ws, size_t ws_size,
                              hipStream_t stream) {
    const float* output      = (const float*)d_in[0];
    const int*   target      = (const int*)d_in[1];
    const int*   longtail    = (const int*)d_in[2];
    const float* loss_weight = (const float*)d_in[3];
    float*       out         = (float*)d_out;

    const int B = in_sizes[1];               // 4096
    const int V = in_sizes[0] / B;           // 32000
    const int L = in_sizes[2] / B;           // 50

    Loss_Longtail_46505905881691_kernel<<<B, 256, 0, stream>>>(
        output, target, longtail, loss_weight, out, V, L);
}